// mean_var_norm_24618752541182
// MI455X (gfx1250) — compile-verified
//
#include <hip/hip_runtime.h>
#include <stdint.h>

#define T_DIM 4000
#define NV4   1000     // T_DIM / 4 float4 vectors per row (16000 B, 16B aligned)
#define BLK   256      // 8 wave32s per row

typedef float fvec4 __attribute__((ext_vector_type(4)));  // native vector for nontemporal b128 stores

__global__ __launch_bounds__(BLK) void mean_var_norm_kernel(
    const float* __restrict__ x,
    const float* __restrict__ lengths,
    float* __restrict__ out)
{
    __shared__ float4 srow[NV4];          // 16000 B row staging buffer
    __shared__ float  wsum[BLK / 32];
    __shared__ float  wsq [BLK / 32];
    __shared__ float  s_mean, s_inv;

    const int f   = blockIdx.x;           // feature index
    const int b   = blockIdx.y;           // batch index (no integer division)
    const int tid = threadIdx.x;
    const size_t row = (size_t)b * gridDim.x + f;

    const float nf = rintf(lengths[b] * (float)T_DIM);   // round-half-even like jnp.round

    const float* grow = x   + row * T_DIM;
    float*       orow = out + row * T_DIM;

    // ---- Phase 1: async copy the whole row global -> LDS (CDNA5 async path) ----
    // Streamed input is read exactly once -> non-temporal hint keeps L2 clean.
#pragma unroll
    for (int k = 0; k < 4; ++k) {
        int idx = tid + k * BLK;
        if (idx < NV4) {
            uint32_t lds_off = (uint32_t)(uintptr_t)(&srow[idx]);     // low 32 bits of generic addr = LDS offset
            uint64_t gaddr   = (uint64_t)(uintptr_t)(grow + idx * 4); // 16B-aligned
            asm volatile("global_load_async_to_lds_b128 %0, %1, off th:TH_LOAD_NT"
                         :: "v"(lds_off), "v"(gaddr) : "memory");
        }
    }
    asm volatile("s_wait_asynccnt 0x0" ::: "memory");   // my wave's async copies landed
    __syncthreads();                                    // everyone's copies landed

    // ---- Phase 2: masked sum / sum-of-squares; keep row in registers ----
    float4 r[4] = {{0.f,0.f,0.f,0.f},{0.f,0.f,0.f,0.f},{0.f,0.f,0.f,0.f},{0.f,0.f,0.f,0.f}};
    float s = 0.f, q = 0.f;
#pragma unroll
    for (int k = 0; k < 4; ++k) {
        int idx = tid + k * BLK;
        if (idx < NV4) {
            float4 v = srow[idx];
            r[k] = v;
            float t0 = (float)(idx * 4);
            float e;
            e = (t0 + 0.f < nf) ? v.x : 0.f; s += e; q = fmaf(e, e, q);
            e = (t0 + 1.f < nf) ? v.y : 0.f; s += e; q = fmaf(e, e, q);
            e = (t0 + 2.f < nf) ? v.z : 0.f; s += e; q = fmaf(e, e, q);
            e = (t0 + 3.f < nf) ? v.w : 0.f; s += e; q = fmaf(e, e, q);
        }
    }

    // wave32 reduction
#pragma unroll
    for (int off = 16; off > 0; off >>= 1) {
        s += __shfl_xor(s, off, 32);
        q += __shfl_xor(q, off, 32);
    }
    const int wid = tid >> 5;
    if ((tid & 31) == 0) { wsum[wid] = s; wsq[wid] = q; }
    __syncthreads();

    if (tid == 0) {
        float S = 0.f, Q = 0.f;
#pragma unroll
        for (int w = 0; w < BLK / 32; ++w) { S += wsum[w]; Q += wsq[w]; }
        float mean = S / nf;
        float var  = fmaxf((Q - S * mean) / (nf - 1.0f), 0.0f);
        float stdv = fmaxf(sqrtf(var), 1e-10f);
        s_mean = mean;
        s_inv  = 1.0f / stdv;
    }
    __syncthreads();

    const float mean = s_mean;
    const float inv  = s_inv;

    // ---- Phase 3: normalize register-resident row, nontemporal b128 stores ----
#pragma unroll
    for (int k = 0; k < 4; ++k) {
        int idx = tid + k * BLK;
        if (idx < NV4) {
            float4 v = r[k];
            fvec4 nv;
            nv.x = (v.x - mean) * inv;
            nv.y = (v.y - mean) * inv;
            nv.z = (v.z - mean) * inv;
            nv.w = (v.w - mean) * inv;
            __builtin_nontemporal_store(nv, (fvec4*)(orow + idx * 4));
        }
    }
}

extern "C" void kernel_launch(void* const* d_in, const int* in_sizes, int n_in,
                              void* d_out, int out_size, void* d_ws, size_t ws_size,
                              hipStream_t stream) {
    const float* x       = (const float*)d_in[0];
    const float* lengths = (const float*)d_in[1];
    float*       out     = (float*)d_out;

    const int B = in_sizes[1];               // 64
    const int F = in_sizes[0] / (B * T_DIM); // 256

    dim3 grid(F, B);                         // blockIdx.x = f, blockIdx.y = b
    mean_var_norm_kernel<<<grid, BLK, 0, stream>>>(x, lengths, out);
}